// HybridModel_24567212933322
// MI455X (gfx1250) — compile-verified
//
#include <hip/hip_runtime.h>

typedef __attribute__((ext_vector_type(2))) float v2f;
typedef __attribute__((ext_vector_type(8))) float v8f;

// ---------------------------------------------------------------------------
// Kernel 1: 4-qubit circuit per 2x2 image block.
// One thread = one block. 16-float statevector kept entirely in registers
// (all loops fully unrolled -> constant indices).
// ---------------------------------------------------------------------------

template <int STR>
__device__ __forceinline__ void ry16(float st[16], float c, float s) {
#pragma unroll
  for (int idx = 0; idx < 16; ++idx) {
    if ((idx & STR) == 0) {
      float s0 = st[idx], s1 = st[idx + STR];
      st[idx]       = c * s0 - s * s1;
      st[idx + STR] = s * s0 + c * s1;
    }
  }
}

template <int CS, int TS>
__device__ __forceinline__ void cnot16(float st[16]) {
#pragma unroll
  for (int idx = 0; idx < 16; ++idx) {
    if ((idx & CS) != 0 && (idx & TS) == 0) {
      float t = st[idx];
      st[idx] = st[idx | TS];
      st[idx | TS] = t;
    }
  }
}

__device__ __forceinline__ void cnot_ring(float st[16]) {
  cnot16<8, 4>(st);  // (0,1)
  cnot16<4, 2>(st);  // (1,2)
  cnot16<2, 1>(st);  // (2,3)
  cnot16<1, 8>(st);  // (3,0)
}

__global__ __launch_bounds__(256) void qcircuit_kernel(
    const float* __restrict__ x, const float* __restrict__ qw,
    float* __restrict__ h, int nblk) {
  int i = blockIdx.x * blockDim.x + threadIdx.x;
  if (i >= nblk) return;

  int b   = i / 196;
  int p   = i - b * 196;
  int i14 = p / 14;
  int j14 = p - i14 * 14;
  const float* xb = x + (size_t)b * 784 + (size_t)(i14 * 2) * 28 + (j14 * 2);

  float ang[4];
  ang[0] = xb[0];
  ang[1] = xb[1];
  ang[2] = xb[28];
  ang[3] = xb[29];

  float st[16];
#pragma unroll
  for (int k = 0; k < 16; ++k) st[k] = 0.f;
  st[0] = 1.f;

  // AngleEmbedding RY merged with entangler-layer-0 RY (same-axis rotations add).
  {
    float s, c;
    __sincosf(0.5f * (ang[0] + qw[0]), &s, &c); ry16<8>(st, c, s);
    __sincosf(0.5f * (ang[1] + qw[1]), &s, &c); ry16<4>(st, c, s);
    __sincosf(0.5f * (ang[2] + qw[2]), &s, &c); ry16<2>(st, c, s);
    __sincosf(0.5f * (ang[3] + qw[3]), &s, &c); ry16<1>(st, c, s);
  }
  cnot_ring(st);
  {
    float s, c;
    __sincosf(0.5f * qw[4], &s, &c); ry16<8>(st, c, s);
    __sincosf(0.5f * qw[5], &s, &c); ry16<4>(st, c, s);
    __sincosf(0.5f * qw[6], &s, &c); ry16<2>(st, c, s);
    __sincosf(0.5f * qw[7], &s, &c); ry16<1>(st, c, s);
  }
  cnot_ring(st);

  float z = 0.f;
#pragma unroll
  for (int idx = 0; idx < 16; ++idx) {
    float p2 = st[idx] * st[idx];
    z += (idx < 8) ? p2 : -p2;
  }
  h[i] = z;
}

// ---------------------------------------------------------------------------
// Kernel 2: out = ReLU(h @ W1 + b1) @ W2 + b2, fused.
// h: (Brows, 196) fp32.  GEMM on the matrix pipe: V_WMMA_F32_16X16X4_F32,
// one wave per 16-row tile, N=30 padded to two 16-wide tiles in LDS.
// ---------------------------------------------------------------------------

__global__ __launch_bounds__(256) void mlp_kernel(
    const float* __restrict__ h, const float* __restrict__ W1,
    const float* __restrict__ b1, const float* __restrict__ W2,
    const float* __restrict__ b2, float* __restrict__ out) {
  __shared__ float w1s[196 * 32];   // W1 padded 30 -> 32 cols
  __shared__ float b1s[32];
  __shared__ float h1s[8][16 * 32]; // per-wave ReLU(h@W1+b1) staging

  const int tid = threadIdx.x;

  for (int idx = tid; idx < 196 * 32; idx += 256) {
    int k = idx >> 5, n = idx & 31;
    w1s[idx] = (n < 30) ? W1[k * 30 + n] : 0.f;
  }
  if (tid < 32) b1s[tid] = (tid < 30) ? b1[tid] : 0.f;
  __syncthreads();

  const int wave = tid >> 5;   // 0..7
  const int lane = tid & 31;
  const int r0   = (blockIdx.x * 8 + wave) * 16;

  const int ln15 = lane & 15;        // A: row M; B/C: col N
  const int half = lane >> 4;        // 0 or 1
  const int kb   = half * 2;         // A/B fragment K sub-offset

  v8f acc0 = {};
  v8f acc1 = {};
  const float* Arow = h + (size_t)(r0 + ln15) * 196;

  for (int k = 0; k < 196; k += 4) {
    v2f a;
    a.x = Arow[k + kb];
    a.y = Arow[k + kb + 1];
    v2f bt0, bt1;
    bt0.x = w1s[(k + kb) * 32 + ln15];
    bt0.y = w1s[(k + kb + 1) * 32 + ln15];
    bt1.x = w1s[(k + kb) * 32 + 16 + ln15];
    bt1.y = w1s[(k + kb + 1) * 32 + 16 + ln15];
    acc0 = __builtin_amdgcn_wmma_f32_16x16x4_f32(
        false, a, false, bt0, (short)0, acc0, false, false);
    acc1 = __builtin_amdgcn_wmma_f32_16x16x4_f32(
        false, a, false, bt1, (short)0, acc1, false, false);
  }

  // bias + ReLU, stage to LDS.  C layout: row = i + 8*half, col = ln15.
  const float bias0 = b1s[ln15];
  const float bias1 = b1s[16 + ln15];
#pragma unroll
  for (int i = 0; i < 8; ++i) {
    int row = i + 8 * half;
    float v0 = acc0[i] + bias0; v0 = v0 > 0.f ? v0 : 0.f;
    float v1 = acc1[i] + bias1; v1 = v1 > 0.f ? v1 : 0.f;
    h1s[wave][row * 32 + ln15]      = v0;
    h1s[wave][row * 32 + 16 + ln15] = v1;
  }
  __syncthreads();

  // Second layer: 16 rows x 2 cols = 32 dot products, one per lane (K=30).
  const int orow = lane & 15;
  const int ocol = lane >> 4;
  float acc = b2[ocol];
#pragma unroll
  for (int k = 0; k < 30; ++k)
    acc += h1s[wave][orow * 32 + k] * W2[k * 2 + ocol];
  out[(size_t)(r0 + orow) * 2 + ocol] = acc;
}

// ---------------------------------------------------------------------------

extern "C" void kernel_launch(void* const* d_in, const int* in_sizes, int n_in,
                              void* d_out, int out_size, void* d_ws, size_t ws_size,
                              hipStream_t stream) {
  const float* x  = (const float*)d_in[0];
  const float* qw = (const float*)d_in[1];
  const float* W1 = (const float*)d_in[2];
  const float* b1 = (const float*)d_in[3];
  const float* W2 = (const float*)d_in[4];
  const float* b2 = (const float*)d_in[5];
  float* out  = (float*)d_out;
  float* hbuf = (float*)d_ws;  // Brows*196 floats (3.2 MB for B=4096)

  const int B    = in_sizes[0] / 784;
  const int nblk = B * 196;

  qcircuit_kernel<<<(nblk + 255) / 256, 256, 0, stream>>>(x, qw, hbuf, nblk);

  const int rowTiles = B / 16;          // B=4096 -> 256 tiles
  mlp_kernel<<<rowTiles / 8, 256, 0, stream>>>(hbuf, W1, b1, W2, b2, out);
}